// HWLRMF_20409684590795
// MI455X (gfx1250) — compile-verified
//
#include <hip/hip_runtime.h>
#include <cmath>

#define BAND 32
#define CHA  128
#define NTOT 4096          // HEI*WID = 64*64
#define ITE  10
#define RNK  3
#define ALPHA 1.05f
#define LSCALE (1.0f/16777216.0f)   // 1/(BAND*CHA*NTOT)

typedef __attribute__((ext_vector_type(2))) float v2f;
typedef __attribute__((ext_vector_type(8))) float v8f;

// ---------------------------------------------------------------- CDNA5 async global->LDS helpers
// GLOBAL_LOAD_ASYNC_TO_LDS_B128: VDST = LDS byte address VGPR, VADDR = 64-bit global address.
// LDS aperture: flat addr[31:0] == LDS byte offset, so truncating a __shared__ pointer is valid.
__device__ __forceinline__ void async_ld_b128(unsigned lds_byte, const float* gptr) {
    asm volatile("global_load_async_to_lds_b128 %0, %1, off"
                 :: "v"(lds_byte), "v"((unsigned long long)(uintptr_t)gptr)
                 : "memory");
}
__device__ __forceinline__ void wait_async8() {
    asm volatile("s_wait_asynccnt 8" ::: "memory");
}
__device__ __forceinline__ void wait_async0() {
    asm volatile("s_wait_asynccnt 0" ::: "memory");
}

// ---------------------------------------------------------------- init
__global__ void hwlrmf_init(float* loss20, float* sumW) {
    int t = threadIdx.x;
    if (t < 20) loss20[t] = 0.f;
    if (t == 0) sumW[0] = 0.f;
}

// ---------------------------------------------------------------- sum(W) stage 1: per-block partials
__global__ __launch_bounds__(256) void hwlrmf_sumw(const float* __restrict__ W,
                                                   float* __restrict__ part, int n) {
    __shared__ float red[256];
    float s = 0.f;
    for (int i = blockIdx.x * 256 + threadIdx.x; i < n; i += gridDim.x * 256) s += W[i];
    red[threadIdx.x] = s;
    __syncthreads();
    for (int o = 128; o > 0; o >>= 1) {
        if (threadIdx.x < o) red[threadIdx.x] += red[threadIdx.x + o];
        __syncthreads();
    }
    if (threadIdx.x == 0) part[blockIdx.x] = red[0];
}

// sum(W) stage 2: deterministic reduce of 1024 partials
__global__ __launch_bounds__(256) void hwlrmf_sumw2(const float* __restrict__ part,
                                                    float* __restrict__ sumW) {
    __shared__ float red[256];
    int t = threadIdx.x;
    red[t] = part[t] + part[t + 256] + part[t + 512] + part[t + 768];
    __syncthreads();
    for (int o = 128; o > 0; o >>= 1) {
        if (t < o) red[t] += red[t + o];
        __syncthreads();
    }
    if (t == 0) sumW[0] = red[0];
}

// ---------------------------------------------------------------- Gram: G[b] = L[b] * L[b]^T
// V_WMMA_F32_16X16X4_F32 + double-buffered GLOBAL_LOAD_ASYNC_TO_LDS_B128 panel staging.
#define KCH   64
#define LDSTR 68    // 16B-aligned rows; spreads fragment reads across banks
#define NCHK  (NTOT / KCH)

__global__ __launch_bounds__(128) void hwlrmf_gram(const float* __restrict__ L,
                                                   float* __restrict__ G) {
    // [buffer][A/B][32 rows * LDSTR]
    __shared__ float pan[2][2][32 * LDSTR];
    const int band = blockIdx.y;
    const int tile = blockIdx.x;            // 16 tiles of 32x32 over the 128x128 G
    const int m0 = (tile >> 2) * 32;
    const int n0 = (tile & 3) * 32;
    const int tid  = threadIdx.x;
    const int wave = tid >> 5;
    const int lane = tid & 31;
    const int wm = (wave >> 1) * 16;        // wave sub-tile inside 32x32
    const int wn = (wave & 1) * 16;
    const float* Lb = L + (size_t)band * CHA * NTOT;

    v8f acc = {};
    const int lrow  = lane & 15;
    const int khalf = (lane >> 4) << 1;     // 0 or 2: K pair selected by half-wave

    // this thread's 4 fixed (row, col4) panel slots
    int rowS[4], c4S[4];
    for (int s = 0; s < 4; ++s) {
        int idx = tid + s * 128;
        rowS[s] = idx >> 4;
        c4S[s]  = (idx & 15) << 2;
    }

    // prologue: async-prefetch chunk 0 into buffer 0 (8 instructions per wave)
    for (int s = 0; s < 4; ++s) {
        async_ld_b128((unsigned)(uintptr_t)&pan[0][0][rowS[s] * LDSTR + c4S[s]],
                      Lb + (size_t)(m0 + rowS[s]) * NTOT + c4S[s]);
        async_ld_b128((unsigned)(uintptr_t)&pan[0][1][rowS[s] * LDSTR + c4S[s]],
                      Lb + (size_t)(n0 + rowS[s]) * NTOT + c4S[s]);
    }

    for (int i = 0; i < NCHK; ++i) {
        const int cur = i & 1;
        if (i + 1 < NCHK) {
            const int kc1 = (i + 1) * KCH;
            const int nxt = cur ^ 1;
            for (int s = 0; s < 4; ++s) {
                async_ld_b128((unsigned)(uintptr_t)&pan[nxt][0][rowS[s] * LDSTR + c4S[s]],
                              Lb + (size_t)(m0 + rowS[s]) * NTOT + kc1 + c4S[s]);
                async_ld_b128((unsigned)(uintptr_t)&pan[nxt][1][rowS[s] * LDSTR + c4S[s]],
                              Lb + (size_t)(n0 + rowS[s]) * NTOT + kc1 + c4S[s]);
            }
            wait_async8();      // 8 prefetches outstanding => current buffer complete
        } else {
            wait_async0();
        }
        __syncthreads();        // all waves' current-buffer loads visible

        const float* arow = &pan[cur][0][(wm + lrow) * LDSTR];
        const float* brow = &pan[cur][1][(wn + lrow) * LDSTR];
        for (int kk = 0; kk < KCH; kk += 4) {
            v2f a, b;
            a.x = arow[kk + khalf]; a.y = arow[kk + khalf + 1];
            b.x = brow[kk + khalf]; b.y = brow[kk + khalf + 1];
            acc = __builtin_amdgcn_wmma_f32_16x16x4_f32(
                false, a, false, b, (short)0, acc, false, false);
        }
        __syncthreads();        // reads done before anyone overwrites this buffer next round
    }

    // C/D layout: VGPR r -> M = r + 8*(lane>=16), N = lane&15
    float* Gb = G + (size_t)band * CHA * CHA;
    const int col   = n0 + wn + lrow;
    const int rbase = m0 + wm + ((lane >> 4) << 3);
    for (int r = 0; r < 8; ++r)
        Gb[(size_t)(rbase + r) * CHA + col] = acc[r];
}

// ---------------------------------------------------------------- top-3 eigenvectors of symmetric G via subspace iteration
__global__ __launch_bounds__(128) void hwlrmf_eig(const float* __restrict__ G,
                                                  float* __restrict__ U) {
    __shared__ float Q[CHA][RNK];
    __shared__ float red[128];
    const int band = blockIdx.x;
    const int t = threadIdx.x;
    const float* Gb = G + (size_t)band * CHA * CHA;

    float q[RNK];
    for (int r = 0; r < RNK; ++r)
        q[r] = Gb[(size_t)t * CHA + r] + ((t == r) ? 1e-3f : 0.f);

    for (int it = 0; it <= 32; ++it) {
        // modified Gram-Schmidt across the 128 rows
        for (int j = 0; j < RNK; ++j) {
            for (int i = 0; i < j; ++i) {
                red[t] = q[i] * q[j];
                __syncthreads();
                for (int o = 64; o > 0; o >>= 1) {
                    if (t < o) red[t] += red[t + o];
                    __syncthreads();
                }
                float d = red[0];
                __syncthreads();
                q[j] -= d * q[i];
            }
            red[t] = q[j] * q[j];
            __syncthreads();
            for (int o = 64; o > 0; o >>= 1) {
                if (t < o) red[t] += red[t + o];
                __syncthreads();
            }
            float inv = rsqrtf(fmaxf(red[0], 1e-30f));
            __syncthreads();
            q[j] *= inv;
        }
        if (it == 32) break;                 // last pass: orthonormalize only
        for (int r = 0; r < RNK; ++r) Q[t][r] = q[r];
        __syncthreads();
        float z[RNK] = {0.f, 0.f, 0.f};
        for (int c = 0; c < CHA; ++c) {
            float g = Gb[(size_t)t * CHA + c];
            for (int r = 0; r < RNK; ++r) z[r] += g * Q[c][r];
        }
        __syncthreads();
        for (int r = 0; r < RNK; ++r) q[r] = z[r];
    }
    for (int r = 0; r < RNK; ++r)
        U[((size_t)band * CHA + t) * 4 + r] = q[r];
}

// ---------------------------------------------------------------- fused projection + losses + L update
__global__ __launch_bounds__(256) void hwlrmf_proj(
    const float* __restrict__ x, const float* __restrict__ W, const float* __restrict__ xg,
    float* __restrict__ L, const float* __restrict__ U, const float* __restrict__ sumW,
    float* __restrict__ uvOut, float* __restrict__ partF, float* __restrict__ partL,
    float rhoCoef, int writeUV)
{
    __shared__ float Us[CHA][RNK];
    __shared__ float redF[256];
    __shared__ float redL[256];
    const int band = blockIdx.y;
    const int t = threadIdx.x;
    const int n = blockIdx.x * 256 + t;
    if (t < CHA)
        for (int r = 0; r < RNK; ++r)
            Us[t][r] = U[((size_t)band * CHA + t) * 4 + r];
    __syncthreads();

    const float rho = sumW[0] * rhoCoef;
    const size_t base = (size_t)band * CHA * NTOT + n;

    // pass 1: utl[r] = sum_c U[c][r] * L[c][n]   (coalesced along n)
    float a0 = 0.f, a1 = 0.f, a2 = 0.f;
    for (int c = 0; c < CHA; ++c) {
        float v = L[base + (size_t)c * NTOT];
        a0 += Us[c][0] * v; a1 += Us[c][1] * v; a2 += Us[c][2] * v;
    }
    // pass 2: uv, losses, L update (in place; this block owns column n)
    float lf = 0.f, ll = 0.f;
    for (int c = 0; c < CHA; ++c) {
        size_t idx = base + (size_t)c * NTOT;
        float uv  = Us[c][0] * a0 + Us[c][1] * a1 + Us[c][2] * a2;
        float xv  = x[idx];
        float wv  = W[idx];
        float d1  = uv - xg[idx];
        float d2  = uv - xv;
        ll += d1 * d1;
        lf += wv * d2 * d2;
        float mu = rho / (wv + rho);
        L[idx] = (1.f - mu) * xv + mu * uv;
        if (writeUV) uvOut[idx] = uv;
    }
    redF[t] = lf; redL[t] = ll;
    __syncthreads();
    for (int o = 128; o > 0; o >>= 1) {
        if (t < o) { redF[t] += redF[t + o]; redL[t] += redL[t + o]; }
        __syncthreads();
    }
    if (t == 0) {
        int bid = band * gridDim.x + blockIdx.x;   // 0..511
        partF[bid] = redF[0];
        partL[bid] = redL[0];
    }
}

// deterministic reduce of the 512 per-block loss partials -> loss_F[t], loss[t]
__global__ __launch_bounds__(256) void hwlrmf_lossfin(const float* __restrict__ partF,
                                                      const float* __restrict__ partL,
                                                      float* __restrict__ lossF,
                                                      float* __restrict__ lossL) {
    __shared__ float redF[256];
    __shared__ float redL[256];
    int t = threadIdx.x;
    redF[t] = partF[t] + partF[t + 256];
    redL[t] = partL[t] + partL[t + 256];
    __syncthreads();
    for (int o = 128; o > 0; o >>= 1) {
        if (t < o) { redF[t] += redF[t + o]; redL[t] += redL[t + o]; }
        __syncthreads();
    }
    if (t == 0) { lossF[0] = redF[0] * LSCALE; lossL[0] = redL[0] * LSCALE; }
}

// ---------------------------------------------------------------- launch
extern "C" void kernel_launch(void* const* d_in, const int* in_sizes, int n_in,
                              void* d_out, int out_size, void* d_ws, size_t ws_size,
                              hipStream_t stream) {
    const float* x  = (const float*)d_in[0];
    const float* W  = (const float*)d_in[1];
    const float* xg = (const float*)d_in[2];
    float* out = (float*)d_out;

    const size_t NE = (size_t)BAND * CHA * NTOT;      // 16,777,216
    float* Lw       = (float*)d_ws;                   // NE floats
    float* Gw       = Lw + NE;                        // BAND*128*128
    float* Uw       = Gw + (size_t)BAND * CHA * CHA;  // BAND*128*4
    float* sumWpart = Uw + (size_t)BAND * CHA * 4;    // 1024
    float* sumW     = sumWpart + 1024;                // 1 (+3 pad)
    float* partF    = sumW + 4;                       // 512
    float* partL    = partF + 512;                    // 512

    float* lossF = out + NE;        // 10
    float* lossL = lossF + ITE;     // 10

    hwlrmf_init<<<1, 32, 0, stream>>>(lossF, sumW);
    hwlrmf_sumw<<<1024, 256, 0, stream>>>(W, sumWpart, (int)NE);
    hwlrmf_sumw2<<<1, 256, 0, stream>>>(sumWpart, sumW);
    hipMemcpyAsync(Lw, x, NE * sizeof(float), hipMemcpyDeviceToDevice, stream);

    float ap = 1.0f;
    for (int t = 0; t < ITE; ++t) {
        hwlrmf_gram<<<dim3(16, BAND), 128, 0, stream>>>(Lw, Gw);
        hwlrmf_eig<<<BAND, 128, 0, stream>>>(Gw, Uw);
        const float rhoCoef = 0.5f * LSCALE * ap;     // rho_t = 0.5*mean(W)*alpha^t
        hwlrmf_proj<<<dim3(NTOT / 256, BAND), 256, 0, stream>>>(
            x, W, xg, Lw, Uw, sumW, out, partF, partL, rhoCoef, (t == ITE - 1) ? 1 : 0);
        hwlrmf_lossfin<<<1, 256, 0, stream>>>(partF, partL, lossF + t, lossL + t);
        ap *= ALPHA;
    }
}